// HausdorffDTLoss_36240934043822
// MI455X (gfx1250) — compile-verified
//
#include <hip/hip_runtime.h>

// HausdorffDTLoss for pred/target [8,1,256,256] fp32.
//   field = EDT(img>0.5) + EDT(img<=0.5), zeroed when sample has no fg
//   loss  = mean((pred-targ)^2 * (field_pred^2 + field_targ^2))
//
// Stage A: per-column 1-D distance scan (fwd+bwd), u16-coded into scratch.
// Stage B: per-row exact parabolic min-plus (O(W^2) VALU), per-row sum.
// Stage C: one-wave WMMA (v_wmma_f32_16x16x4_f32) reduction of row partials.

#define HH 256
#define WW 256
#define BB 8
#define PLANE (HH * WW)          // 65536
#define NPIX  (BB * PLANE)       // 524288
#define INF1D 1e10f
#define THRESH 0.5f

typedef __attribute__((ext_vector_type(2))) float v2f;
typedef __attribute__((ext_vector_type(8))) float v8f;

// ---------------------------------------------------------------------------
// Stage A: vertical 1-D distance (consecutive-run scan, seeded at 1e10 like
// the reference). One block per (batch, src); thread w owns column w.
// Stores min(fwd,bwd) as u16 code (0xFFFF == 1e10 sentinel) for fg and bg.
// Also computes the per-sample "any foreground" flag.
// ---------------------------------------------------------------------------
__global__ void edt_col_kernel(const float* __restrict__ pred,
                               const float* __restrict__ targ,
                               unsigned short* __restrict__ gcode, // [4][BB][PLANE]
                               int* __restrict__ flags)            // [2][BB]
{
    const int b   = blockIdx.x;
    const int src = blockIdx.y;
    const int w   = threadIdx.x;
    const float* img = (src ? targ : pred) + (size_t)b * PLANE;
    unsigned short* gfg = gcode + ((size_t)(src * 2 + 0) * BB + b) * PLANE;
    unsigned short* gbg = gcode + ((size_t)(src * 2 + 1) * BB + b) * PLANE;

    unsigned mbits[HH / 32];
#pragma unroll
    for (int i = 0; i < HH / 32; ++i) mbits[i] = 0u;

    bool any = false;
    float dfg = INF1D, dbg = INF1D;   // fp32 carries, exactly like the reference
    for (int h = 0; h < HH; ++h) {
        float v = img[h * WW + w];
        bool  m = v > THRESH;
        if (m) { mbits[h >> 5] |= (1u << (h & 31)); any = true; }
        dfg = m ? dfg + 1.0f : 0.0f;
        dbg = m ? 0.0f : dbg + 1.0f;
        gfg[h * WW + w] = (dfg >= 1e9f) ? 0xFFFFu : (unsigned short)(int)dfg;
        gbg[h * WW + w] = (dbg >= 1e9f) ? 0xFFFFu : (unsigned short)(int)dbg;
    }
    dfg = INF1D; dbg = INF1D;
    for (int h = HH - 1; h >= 0; --h) {
        bool m = (mbits[h >> 5] >> (h & 31)) & 1u;
        dfg = m ? dfg + 1.0f : 0.0f;
        dbg = m ? 0.0f : dbg + 1.0f;
        unsigned short cf = (dfg >= 1e9f) ? 0xFFFFu : (unsigned short)(int)dfg;
        unsigned short cb = (dbg >= 1e9f) ? 0xFFFFu : (unsigned short)(int)dbg;
        unsigned short of = gfg[h * WW + w];
        unsigned short ob = gbg[h * WW + w];
        gfg[h * WW + w] = cf < of ? cf : of;   // integer min == float min (monotone code)
        gbg[h * WW + w] = cb < ob ? cb : ob;
    }

    __shared__ int sAny;
    if (w == 0) sAny = 0;
    __syncthreads();
    if (any) atomicOr(&sAny, 1);
    __syncthreads();
    if (w == 0) flags[src * BB + b] = sAny;
}

// ---------------------------------------------------------------------------
// Stage B: exact parabolic min along W for all 4 planes of one (batch,row),
// then the per-row contribution to the loss. Fixed-order LDS tree reduction.
// ---------------------------------------------------------------------------
__global__ void edt_row_kernel(const float* __restrict__ pred,
                               const float* __restrict__ targ,
                               const unsigned short* __restrict__ gcode,
                               const int* __restrict__ flags,
                               float* __restrict__ partials)   // [BB*HH]
{
    const int r = blockIdx.x;
    const int b = blockIdx.y;
    const int w = threadIdx.x;

    __shared__ float g2s[4][WW];
    __shared__ float red[WW];

#pragma unroll
    for (int pl = 0; pl < 4; ++pl) {
        unsigned short c = gcode[((size_t)(pl * BB + b)) * PLANE + r * WW + w];
        float g = (c == 0xFFFFu) ? INF1D : (float)c;
        g2s[pl][w] = g * g;
    }
    __syncthreads();

    float d0 = 3.4e38f, d1 = 3.4e38f, d2 = 3.4e38f, d3 = 3.4e38f;
    for (int s = 0; s < WW; ++s) {
        float f  = (float)(w - s);
        float f2 = f * f;                       // exact: |w-s| <= 255
        d0 = fminf(d0, f2 + g2s[0][s]);
        d1 = fminf(d1, f2 + g2s[1][s]);
        d2 = fminf(d2, f2 + g2s[2][s]);
        d3 = fminf(d3, f2 + g2s[3][s]);
    }

    float fp = sqrtf(d0) + sqrtf(d1);           // field = edt(fg)+edt(~fg)
    float ft = sqrtf(d2) + sqrtf(d3);
    if (!flags[b])      fp = 0.0f;              // per-sample no-fg guard
    if (!flags[BB + b]) ft = 0.0f;

    const size_t idx = (size_t)b * PLANE + r * WW + w;
    float e = pred[idx] - targ[idx];
    float v = e * e * (fp * fp + ft * ft);

    red[w] = v;
    __syncthreads();
    for (int s = WW / 2; s > 0; s >>= 1) {
        if (w < s) red[w] += red[w + s];
        __syncthreads();
    }
    if (w == 0) partials[b * HH + r] = red[0];
}

// ---------------------------------------------------------------------------
// Stage C: single-wave WMMA reduction of 2048 row partials.
// With B = ones, sum over ALL elements of D equals 16 * sum(A), independent
// of the A/D lane layouts -> layout-robust matrix-pipe reduction in full fp32.
// ---------------------------------------------------------------------------
__global__ void final_reduce_kernel(const float* __restrict__ partials,
                                    float* __restrict__ out)
{
    const int lane = threadIdx.x;   // 32 threads = one wave, EXEC all ones
    v8f c = {};
    v2f ones; ones.x = 1.0f; ones.y = 1.0f;

    for (int i = 0; i < (BB * HH) / 64; ++i) {   // 32 WMMAs x 64 values
        v2f a;
        a.x = partials[i * 64 + lane];
        a.y = partials[i * 64 + 32 + lane];
        c = __builtin_amdgcn_wmma_f32_16x16x4_f32(
                /*neg_a=*/false, a, /*neg_b=*/false, ones,
                /*c_mod=*/(short)0, c, /*reuse_a=*/false, /*reuse_b=*/false);
    }

    float s = c[0] + c[1] + c[2] + c[3] + c[4] + c[5] + c[6] + c[7];
#pragma unroll
    for (int off = 16; off > 0; off >>= 1)
        s += __shfl_xor(s, off, 32);

    if (lane == 0)
        out[0] = (s * (1.0f / 16.0f)) * (1.0f / (float)NPIX);
}

// ---------------------------------------------------------------------------
extern "C" void kernel_launch(void* const* d_in, const int* in_sizes, int n_in,
                              void* d_out, int out_size, void* d_ws, size_t ws_size,
                              hipStream_t stream)
{
    const float* pred = (const float*)d_in[0];
    const float* targ = (const float*)d_in[1];

    // Workspace layout: 4 MiB of u16 distance codes, 8 KiB partials, 64 B flags.
    unsigned short* gcode = (unsigned short*)d_ws;
    float* partials = (float*)((char*)d_ws + (size_t)4 * BB * PLANE * sizeof(unsigned short));
    int*   flags    = (int*)(partials + BB * HH);
    float* out      = (float*)d_out;

    edt_col_kernel<<<dim3(BB, 2), WW, 0, stream>>>(pred, targ, gcode, flags);
    edt_row_kernel<<<dim3(HH, BB), WW, 0, stream>>>(pred, targ, gcode, flags, partials);
    final_reduce_kernel<<<1, 32, 0, stream>>>(partials, out);
}